// MemoryEfficientNonLinearConv2d_28716151341204
// MI455X (gfx1250) — compile-verified
//
#include <hip/hip_runtime.h>
#include <hip/hip_bf16.h>

// ---------------------------------------------------------------------------
// MemoryEfficientNonLinearConv2d for MI455X (gfx1250, wave32)
//   x:     [4, 32, 32, 32]  f32
//   theta: [64, 32, 3, 3]   f32  -> flattened [64][288], r = cin*9 + kh*3 + kw
//   out:   [4, 64, 32, 32]  f32
// Stage 1: nonlinear "conv" (transcendental-bound VALU loop, LDS-tiled)
// Stage 2: BN batch stats via V_WMMA_F32_16X16X4_F32 ones-matmul reduction
// Stage 3: normalize + clip (in place on d_out)
// ---------------------------------------------------------------------------

typedef __attribute__((ext_vector_type(2))) float v2f;
typedef __attribute__((ext_vector_type(8))) float v8f;

#define B_     4
#define CIN    32
#define HW     32
#define COUT   64
#define RDIM   288          // Cin * 3 * 3
#define NPIX   1024         // 32*32
#define RPAD   289          // odd pad -> conflict-free LDS column access

#define ALPHA_RG   (0.0005625f * 0.1f)   // ALPHA * R_GAIN folded
#define INV_C      (1.0f / 0.075f)       // 1 / (2*n*Vt)
#define VD_OVER_C  (0.1f / 0.075f)
#define BN_EPS_    1e-5f

__device__ __forceinline__ float masked_softplus(float t) {
    // reference: 0 where t <= -20, else log1p(exp(clamp(t,-20,20)))
    float tc = fminf(fmaxf(t, -20.0f), 20.0f);
    float sp = __logf(1.0f + __expf(tc));      // v_exp_f32 / v_log_f32
    return (t > -20.0f) ? sp : 0.0f;
}

// ---------------------------------------------------------------------------
// Kernel 1: out_v[b][co][p] = ALPHA*R_GAIN * sum_r ( sp1^2 - sp2^2 )
// block = (16 pixels, 16 couts) = 256 threads (8 waves)
// grid  = (64 pixel tiles, 4 cout tiles, 4 batches)
// ---------------------------------------------------------------------------
__global__ __launch_bounds__(256)
void nlconv_kernel(const float* __restrict__ x,
                   const float* __restrict__ theta,
                   float* __restrict__ out_v) {
    __shared__ float sTh[16][RPAD];   // theta tile  [co][r]
    __shared__ float sV [16][RPAD];   // im2col tile [pix][r], clipped

    const int tx  = threadIdx.x;            // pixel within tile (0..15)
    const int ty  = threadIdx.y;            // cout  within tile (0..15)
    const int tid = ty * 16 + tx;           // 0..255
    const int p0  = blockIdx.x * 16;        // pixel tile base (0..1008)
    const int co0 = blockIdx.y * 16;        // cout  tile base
    const int b   = blockIdx.z;

    // gfx1250 prefetch of the theta tile (global_prefetch_b8)
    __builtin_prefetch(theta + co0 * RDIM, 0, 0);

    // ---- stage theta tile: 16*288 = 4608 elems, 18 per thread ----
    for (int i = tid; i < 16 * RDIM; i += 256) {
        int co = i / RDIM;
        int r  = i - co * RDIM;
        sTh[co][r] = theta[(co0 + co) * RDIM + r];
    }

    // ---- stage im2col tile with clip(V, 0, 10); zero-pad halo ----
    for (int i = tid; i < 16 * RDIM; i += 256) {
        int p   = i / RDIM;
        int r   = i - p * RDIM;
        int cin = r / 9;
        int k   = r - cin * 9;
        int kh  = k / 3;
        int kw  = k - kh * 3;
        int pl  = p0 + p;
        int oh  = pl >> 5;
        int ow  = pl & 31;
        int ih  = oh + kh - 1;
        int iw  = ow + kw - 1;
        float v = 0.0f;
        if (ih >= 0 && ih < HW && iw >= 0 && iw < HW)
            v = x[((b * CIN + cin) * HW + ih) * HW + iw];
        sV[p][r] = fminf(fmaxf(v, 0.0f), 10.0f);
    }
    __syncthreads();

    // ---- 288-term nonlinear accumulation (transcendental-bound) ----
    float acc = 0.0f;
#pragma unroll 4
    for (int r = 0; r < RDIM; ++r) {
        float diff = sV[tx][r] - sTh[ty][r];
        float t1 = diff * INV_C;
        float t2 = t1 - VD_OVER_C;
        float s1 = masked_softplus(t1);
        float s2 = masked_softplus(t2);
        acc = __builtin_fmaf(s1, s1, acc);
        acc = __builtin_fmaf(-s2, s2, acc);
    }

    out_v[(b * COUT + (co0 + ty)) * NPIX + (p0 + tx)] = acc * ALPHA_RG;
}

// ---------------------------------------------------------------------------
// Kernel 2: per-channel batch stats via WMMA ones-matmul reduction.
// One wave (32 threads) per 16-channel tile; grid = 4 blocks.
// A[16x4] f32: lane L<16 holds A[L,0],A[L,1]; lane L>=16 holds A[L-16,2],A[L-16,3]
// B = ones(4x16)  =>  D[m,n] = row-sum of A, identical across columns n.
// Two accumulators: sum(v) and sum(v^2). Writes scale/shift (gamma,beta folded)
// into d_ws: scale[64] @ ws[0..63], shift[64] @ ws[64..127].
// ---------------------------------------------------------------------------
__global__ __launch_bounds__(32)
void bnstats_wmma_kernel(const float* __restrict__ v,
                         const float* __restrict__ gamma,
                         const float* __restrict__ beta,
                         float* __restrict__ ws) {
    __shared__ float sSum[16];
    __shared__ float sSq [16];

    const int lane = threadIdx.x;          // 0..31
    const int m    = lane & 15;            // channel row within tile
    const int kb   = (lane >> 4) << 1;     // 0 or 2: K sub-column base
    const int c0   = blockIdx.x * 16;      // channel tile base

    const v2f ones = {1.0f, 1.0f};
    v8f accS = {};   // sum(v)
    v8f accQ = {};   // sum(v*v)

    // channel row pointer for this lane's channel (c0+m), walk b and pixels
    for (int b = 0; b < B_; ++b) {
        const float* row = v + (b * COUT + c0 + m) * NPIX;
        for (int pb = 0; pb < NPIX; pb += 4) {
            v2f a = *(const v2f*)(row + pb + kb);         // global_load_b64
            v2f a2 = {a.x * a.x, a.y * a.y};
            accS = __builtin_amdgcn_wmma_f32_16x16x4_f32(
                false, a,  false, ones, (short)0, accS, false, false);
            accQ = __builtin_amdgcn_wmma_f32_16x16x4_f32(
                false, a2, false, ones, (short)0, accQ, false, false);
        }
    }

    // D layout: VGPR j = channel (j + (lane<16 ? 0 : 8)); columns identical.
    if (lane == 0 || lane == 16) {
        int base = (lane >> 4) * 8;
#pragma unroll
        for (int j = 0; j < 8; ++j) {
            sSum[base + j] = accS[j];
            sSq [base + j] = accQ[j];
        }
    }
    __syncthreads();

    if (lane < 16) {
        const float invN = 1.0f / (float)(B_ * NPIX);     // 1/4096
        int   c    = c0 + lane;
        float mean = sSum[lane] * invN;
        float var  = sSq[lane] * invN - mean * mean;      // biased variance
        float rstd = __frsqrt_rn(var + BN_EPS_);
        float scl  = gamma[c] * rstd;
        ws[c]        = scl;                    // scale
        ws[64 + c]   = beta[c] - mean * scl;   // shift
    }
}

// ---------------------------------------------------------------------------
// Kernel 3: in-place normalize + clip.  idx -> (b, c, p) with c = (idx>>10)&63
// ---------------------------------------------------------------------------
__global__ __launch_bounds__(256)
void bnapply_kernel(float* __restrict__ out,
                    const float* __restrict__ ws) {
    int idx = blockIdx.x * 256 + threadIdx.x;
    int c   = (idx >> 10) & (COUT - 1);
    float r = __builtin_fmaf(out[idx], ws[c], ws[64 + c]);
    out[idx] = fminf(fmaxf(r, 0.0f), 10.0f);
}

// ---------------------------------------------------------------------------
extern "C" void kernel_launch(void* const* d_in, const int* in_sizes, int n_in,
                              void* d_out, int out_size, void* d_ws, size_t ws_size,
                              hipStream_t stream) {
    const float* x     = (const float*)d_in[0];
    const float* theta = (const float*)d_in[1];
    const float* gamma = (const float*)d_in[2];
    const float* beta  = (const float*)d_in[3];
    float* out = (float*)d_out;
    float* ws  = (float*)d_ws;   // 128 floats: scale[64], shift[64]

    dim3 g1(NPIX / 16, COUT / 16, B_);   // 64 x 4 x 4 blocks
    dim3 b1(16, 16);
    nlconv_kernel<<<g1, b1, 0, stream>>>(x, theta, out);

    bnstats_wmma_kernel<<<COUT / 16, 32, 0, stream>>>(out, gamma, beta, ws);

    int total = B_ * COUT * NPIX;        // 262144 == out_size
    bnapply_kernel<<<total / 256, 256, 0, stream>>>(out, ws);
}